// SmoothConv_7842610283243
// MI455X (gfx1250) — compile-verified
//
#include <hip/hip_runtime.h>

// ---------------------------------------------------------------------------
// SmoothConv (bilateral point-cloud smoothing), 3 fixed iterations.
// Reformulation: since mask = sign(|p_i|) in {0,1} is idempotent,
//   pts_ker(x)_j == q_t(x+o_j) * mask(x),  q_{t+1} = q_t + imp_{t+1}, q_0 = p,
// so each iteration is a radius-2 stencil on a per-pixel float3 field q, and
// the output is mask * q_3 at the center.
//
// Weights are iteration-invariant. Preferred pipeline (when ws is big enough):
//   pass 0: compute w_j (ONE exp per neighbor) -> 25 j-major planes in ws
//           (52 MB, L2-resident on MI455X's 192 MB L2); also q1 -> d_out
//   pass 1: read cached w_j, q1 -> q2 (ws tail)          [no exp/sqrt at all]
//   pass 2: read cached w_j, q2 -> mask*(q2+imp3) -> d_out
// Fallback (small ws): recompute weights each pass.
// Halo staging uses CDNA5 async global->LDS b64 transfers + s_wait_asynccnt.
// ---------------------------------------------------------------------------

#define TX 32
#define TY 8
#define PADR 2
#define HXX (TX + 2*PADR)          // 36
#define HYY (TY + 2*PADR)          // 12
#define ROWPAIRS (HXX/2)           // 18 b64 transfers per halo row

static constexpr int Himg = 256;
static constexpr int Wimg = 512;

// CDNA5 async global->LDS path (gfx1250). Guarded so either toolchain compiles.
#if defined(__HIP_DEVICE_COMPILE__) && \
    __has_builtin(__builtin_amdgcn_global_load_async_to_lds_b64) && \
    __has_builtin(__builtin_amdgcn_s_wait_asynccnt)
#define ATH_ASYNC 1
#else
#define ATH_ASYNC 0
#endif

#if ATH_ASYNC
// Exact pointee type per compiler diagnostic:
//   __attribute__((__vector_size__(2 * sizeof(int)))) int __device__ *
typedef int ath_v2i __attribute__((vector_size(2 * sizeof(int))));
typedef __attribute__((address_space(1))) ath_v2i* ath_gv2i_p;  // global
typedef __attribute__((address_space(3))) ath_v2i* ath_lv2i_p;  // LDS

__device__ __forceinline__ void ath_async_b64(const float* src, float* dst) {
    __builtin_amdgcn_global_load_async_to_lds_b64(
        (ath_gv2i_p)(void*)src, (ath_lv2i_p)(void*)dst, 0, 0);
}
#endif

// Stage `nplanes` halo planes (3 per field) into LDS. x0 = 32*bx-2 is even, so
// every b64 pair is 8B-aligned and never straddles the x=0 / x=W border.
template <int NPL>
__device__ __forceinline__ void stage_halo(
    float (&smem)[NPL][HYY][HXX],
    const float* f0, const float* f1, const float* f2,  // up to 3 fields
    int bimg, int x0, int y0, int tid)
{
    const int slots = NPL * HYY * ROWPAIRS;
    for (int idx = tid; idx < slots; idx += 256) {
        const int plane = idx / (HYY * ROWPAIRS);
        const int r     = idx % (HYY * ROWPAIRS);
        const int ly    = r / ROWPAIRS;
        const int lx    = (r % ROWPAIRS) * 2;
        const int gy    = y0 + ly;
        const int gx    = x0 + lx;
        const int f = plane / 3, c = plane % 3;
        float* dst = &smem[plane][ly][lx];
        const bool inb = (gy >= 0) && (gy < Himg) && (gx >= 0) && (gx + 1 < Wimg);
        if (inb) {
            const float* sb = (f == 0) ? f0 : ((f == 1) ? f1 : f2);
            const float* src = sb + (((size_t)bimg * 3 + c) * Himg + gy) * (size_t)Wimg + gx;
#if ATH_ASYNC
            ath_async_b64(src, dst);
#else
            *(float2*)dst = *(const float2*)src;
#endif
        } else {
            dst[0] = 0.0f;  // zero padding, matches jax unfold
            dst[1] = 0.0f;
        }
    }
#if ATH_ASYNC
    __builtin_amdgcn_s_wait_asynccnt(0);
#endif
    __syncthreads();
}

// ---- full pass: computes weights in-kernel; optionally caches them ---------
template <bool USE_MASK, bool FINAL, bool STORE_W>
__global__ __launch_bounds__(256) void smooth_pass(
    const float* __restrict__ points,
    const float* __restrict__ normals,
    const float* __restrict__ qin,
    float* __restrict__ qout,
    float* __restrict__ wbuf)   // j-major: wbuf[j*npix + pix]; may be null
{
    __shared__ __align__(16) float smem[9][HYY][HXX];   // p, n, q

    const int bx = blockIdx.x, by = blockIdx.y, bimg = blockIdx.z;
    const int tid = threadIdx.x;
    stage_halo<9>(smem, points, normals, qin, bimg, bx*TX - PADR, by*TY - PADR, tid);

    const int lx = tid & (TX - 1);
    const int ly = tid >> 5;
    const int cx = lx + PADR, cy = ly + PADR;

    float pi[3], ni[3], qi[3], m[3];
#pragma unroll
    for (int c = 0; c < 3; ++c) {
        pi[c] = smem[c][cy][cx];
        ni[c] = smem[3 + c][cy][cx];
        qi[c] = smem[6 + c][cy][cx];
        m[c]  = USE_MASK ? ((pi[c] != 0.0f) ? 1.0f : 0.0f) : 1.0f;
    }

    float d2a[25];
    float maxd2 = 0.0f;
#pragma unroll
    for (int j = 0; j < 25; ++j) {
        const int dy = j / 5 - 2, dx = j % 5 - 2;
        const float a0 = smem[0][cy + dy][cx + dx] - pi[0];
        const float a1 = smem[1][cy + dy][cx + dx] - pi[1];
        const float a2 = smem[2][cy + dy][cx + dx] - pi[2];
        const float d2 = a0 * a0 + a1 * a1 + a2 * a2;
        d2a[j] = d2;
        maxd2 = fmaxf(maxd2, d2);
    }
    const float sigma_c = sqrtf(maxd2) * 0.2f + 1e-5f;
    const float inv2sc2 = 0.5f / (sigma_c * sigma_c);
    const float inv2ss2 = 4.5f;   // 1/(2*(1/3)^2)

    const int gx = bx * TX + lx, gy = by * TY + ly;
    const size_t plane = (size_t)Himg * Wimg;
    const size_t pix   = (size_t)bimg * plane + (size_t)gy * Wimg + gx;
    const size_t npix  = (size_t)gridDim.z * plane;

    float norm = 0.0f;
    float acc[3] = {0.0f, 0.0f, 0.0f};
#pragma unroll
    for (int j = 0; j < 25; ++j) {
        const int dy = j / 5 - 2, dx = j % 5 - 2;
        const float n0 = smem[3][cy + dy][cx + dx];
        const float n1 = smem[4][cy + dy][cx + dx];
        const float n2 = smem[5][cy + dy][cx + dx];
        const float cosang = ni[0] * n0 + ni[1] * n1 + ni[2] * n2;
        const float oc = 1.0f - cosang;
        const float w = __expf(-(d2a[j] * inv2sc2 + oc * oc * inv2ss2));
        if (STORE_W) wbuf[(size_t)j * npix + pix] = w;   // coalesced per j
        norm += w;
        const float dq0 = (smem[6][cy + dy][cx + dx] - qi[0]) * m[0];
        const float dq1 = (smem[7][cy + dy][cx + dx] - qi[1]) * m[1];
        const float dq2 = (smem[8][cy + dy][cx + dx] - qi[2]) * m[2];
        const float s = w * (dq0 * n0 + dq1 * n1 + dq2 * n2);
        acc[0] += n0 * s;
        acc[1] += n1 * s;
        acc[2] += n2 * s;
    }
    const float invn = 1.0f / norm;
    const size_t base = ((size_t)bimg * 3) * plane + (size_t)gy * Wimg + gx;
#pragma unroll
    for (int c = 0; c < 3; ++c) {
        float v = qi[c] + acc[c] * invn;   // q_{t+1}
        if (FINAL) v *= m[c];              // output = mask * q3
        qout[base + c * plane] = v;
    }
}

// ---- cached pass: weights from L2-resident wbuf; no exp/sqrt ---------------
template <bool FINAL>
__global__ __launch_bounds__(256) void smooth_pass_cached(
    const float* __restrict__ points,
    const float* __restrict__ normals,
    const float* __restrict__ qin,
    float* __restrict__ qout,
    const float* __restrict__ wbuf)
{
    __shared__ __align__(16) float smem[6][HYY][HXX];   // n, q only

    const int bx = blockIdx.x, by = blockIdx.y, bimg = blockIdx.z;
    const int tid = threadIdx.x;
    stage_halo<6>(smem, normals, qin, qin, bimg, bx*TX - PADR, by*TY - PADR, tid);

    const int lx = tid & (TX - 1);
    const int ly = tid >> 5;
    const int cx = lx + PADR, cy = ly + PADR;

    const int gx = bx * TX + lx, gy = by * TY + ly;
    const size_t plane = (size_t)Himg * Wimg;
    const size_t pix   = (size_t)bimg * plane + (size_t)gy * Wimg + gx;
    const size_t npix  = (size_t)gridDim.z * plane;
    const size_t base  = ((size_t)bimg * 3) * plane + (size_t)gy * Wimg + gx;

    float ni[3], qi[3], m[3];
#pragma unroll
    for (int c = 0; c < 3; ++c) {
        ni[c] = smem[c][cy][cx];
        qi[c] = smem[3 + c][cy][cx];
        const float pc = points[base + c * plane];       // coalesced center read
        m[c] = (pc != 0.0f) ? 1.0f : 0.0f;               // mask from ORIGINAL p
    }
    (void)ni;

    float norm = 0.0f;
    float acc[3] = {0.0f, 0.0f, 0.0f};
#pragma unroll
    for (int j = 0; j < 25; ++j) {
        const int dy = j / 5 - 2, dx = j % 5 - 2;
        const float w  = wbuf[(size_t)j * npix + pix];   // L2 hit (52 MB < 192 MB)
        const float n0 = smem[0][cy + dy][cx + dx];
        const float n1 = smem[1][cy + dy][cx + dx];
        const float n2 = smem[2][cy + dy][cx + dx];
        norm += w;
        const float dq0 = (smem[3][cy + dy][cx + dx] - qi[0]) * m[0];
        const float dq1 = (smem[4][cy + dy][cx + dx] - qi[1]) * m[1];
        const float dq2 = (smem[5][cy + dy][cx + dx] - qi[2]) * m[2];
        const float s = w * (dq0 * n0 + dq1 * n1 + dq2 * n2);
        acc[0] += n0 * s;
        acc[1] += n1 * s;
        acc[2] += n2 * s;
    }
    const float invn = 1.0f / norm;
#pragma unroll
    for (int c = 0; c < 3; ++c) {
        float v = qi[c] + acc[c] * invn;
        if (FINAL) v *= m[c];
        qout[base + c * plane] = v;
    }
}

extern "C" void kernel_launch(void* const* d_in, const int* in_sizes, int n_in,
                              void* d_out, int out_size, void* d_ws, size_t ws_size,
                              hipStream_t stream) {
    const float* points  = (const float*)d_in[0];
    const float* normals = (const float*)d_in[1];
    float* out = (float*)d_out;              // doubles as q1 scratch
    float* ws  = (float*)d_ws;

    const int B = in_sizes[0] / (3 * Himg * Wimg);
    const size_t npix = (size_t)B * Himg * Wimg;
    dim3 grid(Wimg / TX, Himg / TY, B);      // 16 x 32 x B
    dim3 block(256);                         // 8 wave32 waves

    if (ws_size >= (size_t)28 * npix * sizeof(float)) {
        // weight-cached pipeline: ws = [ w(25 planes) | q2(3 planes) ]
        float* wbuf = ws;
        float* q2   = ws + 25 * npix;
        smooth_pass<false, false, true><<<grid, block, 0, stream>>>(points, normals, points, out, wbuf);
        smooth_pass_cached<false><<<grid, block, 0, stream>>>(points, normals, out, q2,  wbuf);
        smooth_pass_cached<true ><<<grid, block, 0, stream>>>(points, normals, q2,  out, wbuf);
    } else {
        // fallback: recompute weights each pass; ws only holds q2 (3 planes)
        float* q2 = ws;
        smooth_pass<false, false, false><<<grid, block, 0, stream>>>(points, normals, points, out, nullptr);
        smooth_pass<true,  false, false><<<grid, block, 0, stream>>>(points, normals, out, q2,  nullptr);
        smooth_pass<true,  true,  false><<<grid, block, 0, stream>>>(points, normals, q2,  out, nullptr);
    }
}